// ContrastiveLoss_22351009808680
// MI455X (gfx1250) — compile-verified
//
#include <hip/hip_runtime.h>
#include <hip/hip_bf16.h>

// ContrastiveLoss on MI455X (gfx1250):
//  K1: row L2-normalize + per-row sum / sumsq          (512 blocks x 256 thr)
//  K2: Gram matrix via V_WMMA_F32_16X16X4_F32 + fused
//      distance / mask / hinge epilogue + block reduce (128 blocks x 256 thr)
//  K3: deterministic final reduce of 128 partials      (1 block x 128 thr)

#define BATCH   512
#define HALF    256
#define DIM     256
#define PD_EPS  1e-6f
#define INV_T   2.0f      // 1 / TEMPERATURE(=0.5)
#define MARGIN  2.5f

typedef __attribute__((ext_vector_type(2))) float v2f;
typedef __attribute__((ext_vector_type(8))) float v8f;

__device__ __forceinline__ float wave_sum32(float v) {
#pragma unroll
    for (int off = 16; off > 0; off >>= 1)
        v += __shfl_xor(v, off, 32);
    return v;
}

// ---------------- Kernel 1: normalize rows, emit reps / rowsum / rowsq ----
__global__ void cl_normalize_kernel(const float* __restrict__ emb_i,
                                    const float* __restrict__ emb_j,
                                    float* __restrict__ reps,
                                    float* __restrict__ rsum,
                                    float* __restrict__ rsq) {
    const int row = blockIdx.x;          // 0..511
    const int t   = threadIdx.x;         // 0..255 (one element per thread)
    const float* src = (row < HALF) ? (emb_i + (size_t)row * DIM)
                                    : (emb_j + (size_t)(row - HALF) * DIM);
    const float x  = src[t];
    float ss = wave_sum32(x * x);
    float sl = wave_sum32(x);

    __shared__ float red_sq[8];
    __shared__ float red_s[8];
    __shared__ float sh_inv;
    const int wave = t >> 5, lane = t & 31;
    if (lane == 0) { red_sq[wave] = ss; red_s[wave] = sl; }
    __syncthreads();
    if (t == 0) {
        float tot_sq = 0.f, tot_s = 0.f;
#pragma unroll
        for (int w = 0; w < 8; ++w) { tot_sq += red_sq[w]; tot_s += red_s[w]; }
        const float inv = 1.0f / fmaxf(sqrtf(tot_sq), 1e-12f);  // F.normalize eps
        sh_inv   = inv;
        rsum[row] = tot_s  * inv;        // s_a = sum_d z[a,d]
        rsq[row]  = tot_sq * inv * inv;  // q_a = ||z_a||^2
    }
    __syncthreads();
    reps[(size_t)row * DIM + t] = x * sh_inv;
}

// ---------------- Kernel 2: WMMA Gram tile + fused loss epilogue ----------
__global__ void cl_tile_loss_kernel(const float* __restrict__ reps,
                                    const float* __restrict__ rsum,
                                    const float* __restrict__ rsq,
                                    const int*   __restrict__ y,
                                    float* __restrict__ partials) {
    const int t    = threadIdx.x;
    const int wave = t >> 5;
    const int lane = t & 31;
    const int tile = blockIdx.x * 8 + wave;    // 0..1023 (32x32 tiles of 16x16)
    const int tm   = tile >> 5;
    const int tn   = tile & 31;

    // A fragment: 16x4 f32. lanes 0-15 -> M=lane, K={k,k+1};
    //             lanes 16-31 -> M=lane-16, K={k+2,k+3}.  B mirrors with N.
    const int mrow = tm * 16 + (lane & 15);
    const int nrow = tn * 16 + (lane & 15);
    const int koff = (lane >> 4) * 2;
    const float* __restrict__ pa = reps + (size_t)mrow * DIM + koff;
    const float* __restrict__ pb = reps + (size_t)nrow * DIM + koff;

    v8f c = {};
#pragma unroll 8
    for (int k = 0; k < DIM; k += 4) {
        v2f a = *(const v2f*)(pa + k);
        v2f b = *(const v2f*)(pb + k);
        // D = A(16x4) x B(4x16) + C, all f32 — exact-precision Gram tile.
        c = __builtin_amdgcn_wmma_f32_16x16x4_f32(
                /*neg_a=*/false, a, /*neg_b=*/false, b,
                /*c_mod=*/(short)0, c, /*reuse_a=*/false, /*reuse_b=*/false);
    }

    // Epilogue: lane holds C VGPR r => global row a = tm*16 + r + 8*(lane>>4),
    // col b = tn*16 + (lane&15).
    const int   nb = tn * 16 + (lane & 15);
    const float sB = rsum[nb];
    const float qB = rsq[nb];
    const int   yB = y[nb];
    float local = 0.0f;
#pragma unroll
    for (int r = 0; r < 8; ++r) {
        const int na = tm * 16 + r + ((lane >> 4) << 3);
        if (na != nb) {                              // off-diagonal only
            const float g = c[r];
            float dist2 = rsq[na] + qB - 2.0f * g
                        + 2.0f * PD_EPS * (rsum[na] - sB)
                        + (float)DIM * PD_EPS * PD_EPS;
            dist2 = fmaxf(dist2, 0.0f);
            const float dd = sqrtf(dist2) * INV_T;   // dist / temperature
            if (y[na] == yB) {
                local += dd * dd;                    // positive pair
            } else {
                const float far = fmaxf(MARGIN - dd, 0.0f);
                local += far * far;                  // negative pair (hinge)
            }
        }
    }

    // Deterministic block reduction: wave shfl -> LDS -> thread 0.
    local = wave_sum32(local);
    __shared__ float red[8];
    if (lane == 0) red[wave] = local;
    __syncthreads();
    if (t == 0) {
        float tot = 0.f;
#pragma unroll
        for (int w = 0; w < 8; ++w) tot += red[w];
        partials[blockIdx.x] = tot;
    }
}

// ---------------- Kernel 3: final reduce (fixed order => deterministic) ---
__global__ void cl_final_reduce_kernel(const float* __restrict__ partials,
                                       float* __restrict__ out) {
    const int t = threadIdx.x;              // 128 threads, 4 waves
    float v = partials[t];
    v = wave_sum32(v);
    __shared__ float red[4];
    const int wave = t >> 5, lane = t & 31;
    if (lane == 0) red[wave] = v;
    __syncthreads();
    if (t == 0) {
        const float tot = red[0] + red[1] + red[2] + red[3];
        out[0] = tot / (2.0f * (float)BATCH);
    }
}

extern "C" void kernel_launch(void* const* d_in, const int* in_sizes, int n_in,
                              void* d_out, int out_size, void* d_ws, size_t ws_size,
                              hipStream_t stream) {
    (void)in_sizes; (void)n_in; (void)out_size; (void)ws_size;
    const float* emb_i = (const float*)d_in[0];
    const float* emb_j = (const float*)d_in[1];
    const int*   y     = (const int*)d_in[2];    // jax default: int64 demoted to int32

    float* reps     = (float*)d_ws;              // 512*256 f32
    float* rsum     = reps + (size_t)BATCH * DIM; // 512
    float* rsq      = rsum + BATCH;               // 512
    float* partials = rsq  + BATCH;               // 128

    cl_normalize_kernel<<<BATCH, DIM, 0, stream>>>(emb_i, emb_j, reps, rsum, rsq);
    cl_tile_loss_kernel<<<128, 256, 0, stream>>>(reps, rsum, rsq, y, partials);
    cl_final_reduce_kernel<<<1, 128, 0, stream>>>(partials, (float*)d_out);
}